// SmoothCondition_91199335563439
// MI455X (gfx1250) — compile-verified
//
#include <hip/hip_runtime.h>
#include <hip/hip_bf16.h>

typedef __attribute__((ext_vector_type(2))) float v2f;
typedef __attribute__((ext_vector_type(8))) float v8f;

#define KCHUNK 32
#define NEG_INF_F (-1e9f)

static __device__ __forceinline__ v8f wmma_f32x4(v2f a, v2f b, v8f c) {
  // D = A(16x4) * B(4x16) + C, exact f32 matrix pipe on CDNA5
  return __builtin_amdgcn_wmma_f32_16x16x4_f32(
      /*neg_a=*/false, a, /*neg_b=*/false, b,
      /*c_mod=*/(short)0, c, /*reuse_a=*/false, /*reuse_b=*/false);
}

// ---------------------------------------------------------------------------
// sq[b,s] = sum_c x^2   (also zero-inits the weighted accumulator)
// ---------------------------------------------------------------------------
__global__ __launch_bounds__(256) void sq_kernel(const float* __restrict__ x,
                                                 float* __restrict__ sq,
                                                 float* __restrict__ weighted,
                                                 int C) {
  int row  = blockIdx.x * 8 + (threadIdx.x >> 5);
  int lane = threadIdx.x & 31;
  const float4* xr = (const float4*)(x + (size_t)row * C);
  float acc = 0.f;
  for (int i = lane; i < (C >> 2); i += 32) {
    float4 v = xr[i];
    acc += v.x * v.x + v.y * v.y + v.z * v.z + v.w * v.w;
  }
#pragma unroll
  for (int m = 1; m < 32; m <<= 1) acc += __shfl_xor(acc, m, 32);
  if (lane == 0) {
    sq[row] = acc;
    weighted[row] = 0.f;
  }
}

// ---------------------------------------------------------------------------
// logits[r] = u . tanh(x[r,:] W + b)   for 16 rows per workgroup.
// 8 waves, wave w computes the 16x16 tile of y = xW over columns [16w,16w+16).
// ---------------------------------------------------------------------------
__global__ __launch_bounds__(256) void logits_kernel(const float* __restrict__ x,
                                                     const float* __restrict__ W,
                                                     const float* __restrict__ bvec,
                                                     const float* __restrict__ uvec,
                                                     float* __restrict__ logits,
                                                     int C, int A) {
  __shared__ float Xt[16][KCHUNK + 2];     // 16 rows x 32 K
  __shared__ float Wt[KCHUNK][128 + 2];    // 32 K x 128 A
  __shared__ float partial[16][8];

  const int t    = threadIdx.x;
  const int wv   = t >> 5;       // wave 0..7
  const int lane = t & 31;
  const int half = lane >> 4;    // 0: lanes 0-15, 1: lanes 16-31
  const int l16  = lane & 15;
  const int r0   = blockIdx.x * 16;      // flat (b*S + s) row base
  const int a0   = wv * 16;

  v8f acc = {};
  const float* xb = x + (size_t)r0 * C;

  for (int k0 = 0; k0 < C; k0 += KCHUNK) {
#pragma unroll
    for (int it = 0; it < 2; ++it) {     // 16*32 = 512 elems
      int e = t + it * 256;
      int r = e >> 5, c = e & 31;
      Xt[r][c] = xb[(size_t)r * C + k0 + c];
    }
#pragma unroll
    for (int it = 0; it < 16; ++it) {    // 32*128 = 4096 elems
      int e = t + it * 256;
      int kk = e >> 7, a = e & 127;
      Wt[kk][a] = W[(size_t)(k0 + kk) * A + a];
    }
    __syncthreads();
#pragma unroll
    for (int kk = 0; kk < KCHUNK; kk += 4) {
      v2f af, bf;
      af.x = Xt[l16][kk + half * 2];
      af.y = Xt[l16][kk + half * 2 + 1];
      bf.x = Wt[kk + half * 2][a0 + l16];
      bf.y = Wt[kk + half * 2 + 1][a0 + l16];
      acc = wmma_f32x4(af, bf, acc);
    }
    __syncthreads();
  }

  // epilogue: tanh(y + b) * u, reduce over this wave's 16 columns
  const int a = a0 + l16;
  const float bb = bvec[a];
  const float uu = uvec[a];
  float sums[8];
#pragma unroll
  for (int v = 0; v < 8; ++v) {
    float y = tanhf(acc[v] + bb) * uu;
#pragma unroll
    for (int m = 1; m < 16; m <<= 1) y += __shfl_xor(y, m, 16);
    sums[v] = y;
  }
  if (l16 == 0) {
#pragma unroll
    for (int v = 0; v < 8; ++v) partial[v + half * 8][wv] = sums[v];
  }
  __syncthreads();
  if (t < 16) {
    float s = 0.f;
#pragma unroll
    for (int w8 = 0; w8 < 8; ++w8) s += partial[t][w8];
    logits[r0 + t] = s;
  }
}

// ---------------------------------------------------------------------------
// masked softmax over S per batch
// ---------------------------------------------------------------------------
__global__ __launch_bounds__(512) void softmax_kernel(const float* __restrict__ logits,
                                                      const int* __restrict__ lens,
                                                      float* __restrict__ score,
                                                      int S) {
  __shared__ float red[512];
  const int b = blockIdx.x;
  const int s = threadIdx.x;
  float l = logits[b * S + s];
  l = (s < lens[b]) ? l : NEG_INF_F;
  red[s] = l;
  __syncthreads();
  for (int off = 256; off > 0; off >>= 1) {
    if (s < off) red[s] = fmaxf(red[s], red[s + off]);
    __syncthreads();
  }
  const float mx = red[0];
  __syncthreads();
  const float e = __expf(l - mx);
  red[s] = e;
  __syncthreads();
  for (int off = 256; off > 0; off >>= 1) {
    if (s < off) red[s] += red[s + off];
    __syncthreads();
  }
  score[b * S + s] = e / red[0];
}

// ---------------------------------------------------------------------------
// gram tile: WG computes 64x128 of x.x^T, each wave a 32x32 block (2x2 tiles
// of 16x16 -> 4 v8f accumulators, 2 LDS dwords per WMMA). Fused RBF epilogue:
// weighted[b,i] += sum_j exp(-max(sq_i+sq_j-2*gram_ij,0)/2) * score[b,j]
// ---------------------------------------------------------------------------
__global__ __launch_bounds__(256) void gram_kernel(const float* __restrict__ x,
                                                   const float* __restrict__ sq,
                                                   const float* __restrict__ score,
                                                   float* __restrict__ weighted,
                                                   int S, int C) {
  __shared__ float Xi[64][KCHUNK + 2];
  __shared__ float Xj[128][KCHUNK + 2];

  const int t    = threadIdx.x;
  const int wv   = t >> 5;
  const int lane = t & 31;
  const int half = lane >> 4;
  const int l16  = lane & 15;
  const int wi2  = wv >> 2;          // 0..1  (32-row block inside WG)
  const int wj2  = wv & 3;           // 0..3  (32-col block inside WG)

  const int b  = blockIdx.z;
  const int i0 = blockIdx.y * 64;
  const int j0 = blockIdx.x * 128;
  const float* xb = x + (size_t)b * S * C;

  v8f acc[2][2] = {};
  for (int k0 = 0; k0 < C; k0 += KCHUNK) {
#pragma unroll
    for (int it = 0; it < 8; ++it) {   // 64*32 = 2048 elems
      int e = t + it * 256;
      int r = e >> 5, c = e & 31;
      Xi[r][c] = xb[(size_t)(i0 + r) * C + k0 + c];
    }
#pragma unroll
    for (int it = 0; it < 16; ++it) {  // 128*32 = 4096 elems
      int e = t + it * 256;
      int r = e >> 5, c = e & 31;
      Xj[r][c] = xb[(size_t)(j0 + r) * C + k0 + c];
    }
    // prefetch next chunk: one byte per row pulls the whole 128B line that
    // holds the next 32 floats of that row (k0 is 128B aligned)
    if (k0 + KCHUNK < C) {
      if (t < 64) {
        __builtin_prefetch(&xb[(size_t)(i0 + t) * C + k0 + KCHUNK], 0, 3);
      } else if (t < 192) {
        __builtin_prefetch(&xb[(size_t)(j0 + (t - 64)) * C + k0 + KCHUNK], 0, 3);
      }
    }
    __syncthreads();
#pragma unroll
    for (int kk = 0; kk < KCHUNK; kk += 4) {
      const int col = kk + half * 2;
      v2f a0, a1, b0, b1;
      a0.x = Xi[wi2 * 32 + l16][col];       a0.y = Xi[wi2 * 32 + l16][col + 1];
      a1.x = Xi[wi2 * 32 + 16 + l16][col];  a1.y = Xi[wi2 * 32 + 16 + l16][col + 1];
      b0.x = Xj[wj2 * 32 + l16][col];       b0.y = Xj[wj2 * 32 + l16][col + 1];
      b1.x = Xj[wj2 * 32 + 16 + l16][col];  b1.y = Xj[wj2 * 32 + 16 + l16][col + 1];
      acc[0][0] = wmma_f32x4(a0, b0, acc[0][0]);
      acc[0][1] = wmma_f32x4(a0, b1, acc[0][1]);
      acc[1][0] = wmma_f32x4(a1, b0, acc[1][0]);
      acc[1][1] = wmma_f32x4(a1, b1, acc[1][1]);
    }
    __syncthreads();
  }

  // fused RBF epilogue; tj=0/1 share the same output row -> combine before
  // the cross-lane reduction (halves the atomic count)
  const float* sqb = sq + (size_t)b * S;
  const int jb = j0 + wj2 * 32;
  const float sqj0 = sqb[jb + l16];
  const float sqj1 = sqb[jb + 16 + l16];
  const float scj0 = score[(size_t)b * S + jb + l16];
  const float scj1 = score[(size_t)b * S + jb + 16 + l16];
#pragma unroll
  for (int ti = 0; ti < 2; ++ti) {
#pragma unroll
    for (int v = 0; v < 8; ++v) {
      const int i = i0 + wi2 * 32 + ti * 16 + v + half * 8;
      const float sqi = sqb[i];
      float d0 = fmaxf(sqi + sqj0 - 2.f * acc[ti][0][v], 0.f);
      float d1 = fmaxf(sqi + sqj1 - 2.f * acc[ti][1][v], 0.f);
      float wgt = __expf(-0.5f * d0) * scj0 + __expf(-0.5f * d1) * scj1;
#pragma unroll
      for (int m = 1; m < 16; m <<= 1) wgt += __shfl_xor(wgt, m, 16);
      if (l16 == 0) atomicAdd(&weighted[(size_t)b * S + i], wgt);
    }
  }
}

// ---------------------------------------------------------------------------
// out = min(x + onehot(target) * weighted, 1.0), float4 streaming
// ---------------------------------------------------------------------------
__global__ __launch_bounds__(256) void out_kernel(const float* __restrict__ x,
                                                  const float* __restrict__ weighted,
                                                  const int* __restrict__ target,
                                                  float* __restrict__ out,
                                                  int B, int S, int C) {
  const size_t total4 = (size_t)B * S * C / 4;
  const size_t stride = (size_t)gridDim.x * blockDim.x;
  const float4* x4 = (const float4*)x;
  float4* o4 = (float4*)out;
  for (size_t idx = (size_t)blockIdx.x * blockDim.x + threadIdx.x; idx < total4;
       idx += stride) {
    const size_t base = idx * 4;
    const int c = (int)(base % (size_t)C);
    const size_t row = base / (size_t)C;     // flat b*S + s
    const int b = (int)(row / (size_t)S);
    float4 v = x4[idx];
    const int tc = target[b];
    if (tc >= c && tc < c + 4) {
      ((float*)&v)[tc - c] += weighted[row];
    }
    v.x = fminf(v.x, 1.f);
    v.y = fminf(v.y, 1.f);
    v.z = fminf(v.z, 1.f);
    v.w = fminf(v.w, 1.f);
    o4[idx] = v;
  }
}

// ---------------------------------------------------------------------------
extern "C" void kernel_launch(void* const* d_in, const int* in_sizes, int n_in,
                              void* d_out, int out_size, void* d_ws, size_t ws_size,
                              hipStream_t stream) {
  const float* x      = (const float*)d_in[0];
  const int*   lens   = (const int*)d_in[1];
  const int*   target = (const int*)d_in[2];
  const float* W      = (const float*)d_in[3];
  const float* bvec   = (const float*)d_in[4];
  const float* uvec   = (const float*)d_in[5];
  float* out = (float*)d_out;

  const int B = 32, S = 512, C = 2048, A = 128;
  const int R = B * S;  // 16384 flat rows

  float* ws       = (float*)d_ws;
  float* logits   = ws;          // R floats
  float* score    = ws + R;      // R floats
  float* sqbuf    = ws + 2 * R;  // R floats
  float* weighted = ws + 3 * R;  // R floats (zero-initialized by sq_kernel)

  sq_kernel<<<R / 8, 256, 0, stream>>>(x, sqbuf, weighted, C);
  logits_kernel<<<R / 16, 256, 0, stream>>>(x, W, bvec, uvec, logits, C, A);
  softmax_kernel<<<B, S, 0, stream>>>(logits, lens, score, S);
  gram_kernel<<<dim3(S / 128, S / 64, B), 256, 0, stream>>>(x, sqbuf, score,
                                                            weighted, S, C);
  out_kernel<<<4096, 256, 0, stream>>>(x, weighted, target, out, B, S, C);
}